// NonLocalDenoiser_44633300140716
// MI455X (gfx1250) — compile-verified
//
#include <hip/hip_runtime.h>

// ---------------------------------------------------------------------------
// NonLocalDenoiser forward for MI455X (gfx1250, wave32, WMMA).
//
// Every bilinear layer  Z = Wh * X * Wv^T + b  runs as ONE fused kernel per
// patch, templated on compile-time dims so the WMMA k-loops fully unroll:
//   stage 1: X staged in zero-padded LDS -> Y = X*Wv^T via V_WMMA_F32_16X16X4_F32
//   stage 2: Z = Wh*Y (+bias, relu) from LDS -> global (strided concat writes)
// Edge guards use clamped-index loads * float mask (v_cndmask), never
// exec-predicated branches.  BN stats / fold / unfold are scalar kernels.
// ---------------------------------------------------------------------------

typedef __attribute__((ext_vector_type(2))) float v2f;
typedef __attribute__((ext_vector_type(8))) float v8f;

#define IPN   2
#define PPN   9216
#define IPT   (IPN * PPN)      // 18432 patches total
#define PNE   75               // patch_numel
#define VSZ   80               // ver_size
#define EPSB  1e-5f

__device__ __forceinline__ int imin_(int a, int b) { return a < b ? a : b; }

// ---------------------------------------------------------------------------
// Fused ver+hor bilinear layer, compile-time shapes.
//   X   : [IPT, inHorTot, K]   (rows inHorOff .. inHorOff+HIN-1 used)
//   Wv  : [NW, K]    Wh : [GOUT, HIN]    bias : [GOUT, NW]
//   Out : [IPT, outHorTot, NW] (rows outHorOff .. outHorOff+GOUT-1 written)
// WMMA f32 16x16x4 fragment layouts per CDNA5 ISA 7.12.2:
//   A (16x4): lane<16 -> rows M=lane, K = k0,k0+1 ; lane>=16 -> K = k0+2,k0+3
//   B (4x16): vgpr0 lanes0-15 row k0, lanes16-31 row k0+2; vgpr1 -> k0+1/k0+3
//   C/D     : vgpr j -> M = 16*tm + j + 8*(lane>>4), N = lane&15
// ---------------------------------------------------------------------------
template <int HIN, int K, int NW, int GOUT, int RELU>
__global__ __launch_bounds__(256)
void verhor_t(const float* __restrict__ X, int inHorTot, int inHorOff,
              const float* __restrict__ Wv,
              const float* __restrict__ Wh,
              const float* __restrict__ bias,
              float* __restrict__ Out, int outHorTot, int outHorOff)
{
    constexpr int TM1  = (HIN + 15) / 16;
    constexpr int TN   = (NW + 15) / 16;
    constexpr int TM2  = (GOUT + 15) / 16;
    constexpr int KS1  = (K + 3) / 4;
    constexpr int KS2  = (HIN + 3) / 4;
    constexpr int HINP = TM1 * 16;          // padded X rows
    constexpr int KP2  = KS1 * 4 + 2;       // padded X cols (even -> b64 aligned)
    constexpr int NWP  = TN * 16 + 1;       // padded Y cols

    __shared__ float Xs[HINP][KP2];         // zero-padded input tile
    __shared__ float Ys[HINP][NWP];         // stage-1 result (zero-padded tiles)

    const int p    = blockIdx.x;
    const int tid  = threadIdx.x;
    const int lane = tid & 31;
    const int wvid = tid >> 5;              // 0..7 (wave32)
    const int hs   = lane >> 4;             // half-select within wave
    const int l15  = lane & 15;

    const float* Xp = X + ((size_t)p * inHorTot + inHorOff) * (size_t)K;

    // ---- cooperative zero-padded staging of X (clamped loads, no branches)
    for (int idx = tid; idx < HINP * KP2; idx += 256) {
        const int r = idx / KP2, cix = idx % KP2;
        const float m = (r < HIN && cix < K) ? 1.f : 0.f;
        Xs[r][cix] = Xp[imin_(r, HIN - 1) * K + imin_(cix, K - 1)] * m;
    }
    __syncthreads();

    // ---- stage 1: Y = X * Wv^T -------------------------------------------
    {
        const int   n    = l15;             // per-tile column offset added below
        const int   ncl0 = (NW % 16) ? 0 : 1;  // 1 => no column mask needed
        for (int t = wvid; t < TM1 * TN; t += 8) {
            const int tm = t / TN, tn = t % TN;
            const int row = tm * 16 + l15;
            const int nn  = tn * 16 + n;
            const int ncl = ncl0 ? nn : imin_(nn, NW - 1);
            const float bm = (ncl0 || nn < NW) ? 1.f : 0.f;
            v8f c = {0.f, 0.f, 0.f, 0.f, 0.f, 0.f, 0.f, 0.f};
            #pragma unroll
            for (int ks = 0; ks < KS1; ++ks) {
                const int kb = ks * 4 + hs * 2;
                v2f a = *(const v2f*)&Xs[row][kb];        // ds_load_b64
                v2f b;
                const float m0 = (kb     < K) ? bm : 0.f; // folds except last ks
                const float m1 = (kb + 1 < K) ? bm : 0.f;
                b.x = Wv[ncl * K + imin_(kb,     K - 1)] * m0;
                b.y = Wv[ncl * K + imin_(kb + 1, K - 1)] * m1;
                c = __builtin_amdgcn_wmma_f32_16x16x4_f32(false, a, false, b,
                                                          (short)0, c, false, false);
            }
            #pragma unroll
            for (int j = 0; j < 8; ++j)
                Ys[tm * 16 + j + hs * 8][nn] = c[j];
        }
    }
    __syncthreads();

    // ---- stage 2: Z = Wh * Y + bias --------------------------------------
    float* Op = Out + ((size_t)p * outHorTot + outHorOff) * (size_t)NW;
    for (int t = wvid; t < TM2 * TN; t += 8) {
        const int tg = t / TN, tn = t % TN;
        const int g  = tg * 16 + l15;
        const int n  = tn * 16 + l15;
        const int gcl = imin_(g, GOUT - 1);
        const float gm = (g < GOUT) ? 1.f : 0.f;
        v8f c = {0.f, 0.f, 0.f, 0.f, 0.f, 0.f, 0.f, 0.f};
        #pragma unroll
        for (int ks = 0; ks < KS2; ++ks) {
            const int kb = ks * 4 + hs * 2;
            v2f a, b;
            const float m0 = (kb     < HIN) ? gm : 0.f;
            const float m1 = (kb + 1 < HIN) ? gm : 0.f;
            a.x = Wh[gcl * HIN + imin_(kb,     HIN - 1)] * m0;
            a.y = Wh[gcl * HIN + imin_(kb + 1, HIN - 1)] * m1;
            b.x = Ys[imin_(kb,     HINP - 1)][n];          // rows >= HIN are zeros
            b.y = Ys[imin_(kb + 1, HINP - 1)][n];
            c = __builtin_amdgcn_wmma_f32_16x16x4_f32(false, a, false, b,
                                                      (short)0, c, false, false);
        }
        #pragma unroll
        for (int j = 0; j < 8; ++j) {
            const int m = tg * 16 + j + hs * 8;
            if (m < GOUT && n < NW) {                      // store guard only
                float v = c[j] + bias[m * NW + n];
                if (RELU) v = fmaxf(v, 0.f);
                Op[(size_t)m * NW + n] = v;
            }
        }
    }
}

// ---------------------------------------------------------------------------
// BatchNorm helpers (training-mode stats over (i, p, ver) per hor channel).
// ---------------------------------------------------------------------------
__global__ void zero_kernel(float* __restrict__ p, int n)
{
    int i = blockIdx.x * blockDim.x + threadIdx.x;
    if (i < n) p[i] = 0.f;
}

__global__ __launch_bounds__(256)
void bn_stats_kernel(const float* __restrict__ T, int C, int Nw,
                     float* __restrict__ stats /* [C][2] */)
{
    const int c = blockIdx.x;
    __shared__ float ss[256], sq[256];
    float s = 0.f, q = 0.f;
    const size_t total = (size_t)IPT * Nw;
    for (size_t idx = (size_t)blockIdx.y * blockDim.x + threadIdx.x; idx < total;
         idx += (size_t)gridDim.y * blockDim.x) {
        const size_t p = idx / Nw;
        const int    w = (int)(idx % Nw);
        const float  v = T[(p * C + c) * Nw + w];
        s += v; q += v * v;
    }
    ss[threadIdx.x] = s; sq[threadIdx.x] = q;
    __syncthreads();
    for (int o = 128; o > 0; o >>= 1) {
        if ((int)threadIdx.x < o) {
            ss[threadIdx.x] += ss[threadIdx.x + o];
            sq[threadIdx.x] += sq[threadIdx.x + o];
        }
        __syncthreads();
    }
    if (threadIdx.x == 0) {
        atomicAdd(&stats[2 * c],     ss[0]);
        atomicAdd(&stats[2 * c + 1], sq[0]);
    }
}

__global__ void bn_finalize_kernel(const float* __restrict__ stats,
                                   const float* __restrict__ gamma,
                                   const float* __restrict__ beta,
                                   int C, float invCnt,
                                   float* __restrict__ scl, float* __restrict__ shf)
{
    int c = threadIdx.x;
    if (c < C) {
        float mu  = stats[2 * c] * invCnt;
        float var = stats[2 * c + 1] * invCnt - mu * mu;
        float sc  = gamma[c] * rsqrtf(var + EPSB);
        scl[c] = sc;
        shf[c] = beta[c] - sc * mu;
    }
}

__global__ void bnrelu_kernel(const float* __restrict__ T,
                              const float* __restrict__ scl,
                              const float* __restrict__ shf,
                              int C, int Nw, float* __restrict__ Out, size_t total)
{
    size_t i = (size_t)blockIdx.x * blockDim.x + threadIdx.x;
    if (i >= total) return;
    int c = (int)((i / Nw) % C);
    Out[i] = fmaxf(scl[c] * T[i] + shf[c], 0.f);
}

// out = a*H + (1-a)*relu(bn(T)), written with hor-stride into a concat buffer
__global__ void combine_kernel(const float* __restrict__ H, const float* __restrict__ T,
                               const float* __restrict__ scl, const float* __restrict__ shf,
                               const float* __restrict__ aP, int C, int Nw,
                               float* __restrict__ Out, int outHorTot, int outHorOff,
                               size_t total)
{
    size_t i = (size_t)blockIdx.x * blockDim.x + threadIdx.x;
    if (i >= total) return;
    const int    w = (int)(i % Nw);
    const size_t r = i / Nw;
    const int    c = (int)(r % C);
    const size_t p = r / C;
    const float  a = aP[0];
    const float h2 = fmaxf(scl[c] * T[i] + shf[c], 0.f);
    Out[(p * outHorTot + outHorOff + c) * (size_t)Nw + w] = a * H[i] + (1.f - a) * h2;
}

// ---------------------------------------------------------------------------
// agg0: fold (k=5,d=1,H=W=100) / count, then unfold.  Direct gather form.
// ---------------------------------------------------------------------------
__global__ void fold0_kernel(const float* __restrict__ ypre, float* __restrict__ F)
{
    const int TOT = IPN * 3 * 100 * 100;
    int idx = blockIdx.x * blockDim.x + threadIdx.x;
    if (idx >= TOT) return;
    const int x = idx % 100;
    const int y = (idx / 100) % 100;
    const int c = (idx / 10000) % 3;
    const int i = idx / 30000;
    const int ky0 = max(0, y - 95), ky1 = min(4, y);
    const int kx0 = max(0, x - 95), kx1 = min(4, x);
    float s = 0.f;
    for (int ky = ky0; ky <= ky1; ++ky)
        for (int kx = kx0; kx <= kx1; ++kx) {
            const int p = (y - ky) * 96 + (x - kx);
            s += ypre[((size_t)i * PPN + p) * PNE + c * 25 + ky * 5 + kx];
        }
    F[idx] = s / (float)((ky1 - ky0 + 1) * (kx1 - kx0 + 1));
}

__global__ void unfold0_kernel(const float* __restrict__ F, float* __restrict__ yagg)
{
    const int TOT = IPN * PPN * PNE;
    int idx = blockIdx.x * blockDim.x + threadIdx.x;
    if (idx >= TOT) return;
    const int e = idx % PNE;
    const int p = (idx / PNE) % PPN;
    const int i = idx / (PNE * PPN);
    const int c = e / 25, ky = (e / 5) % 5, kx = e % 5;
    const int py = p / 96, px = p % 96;
    yagg[idx] = F[((i * 3 + c) * 100 + py + ky) * 100 + px + kx];
}

// ---------------------------------------------------------------------------
// agg1: fold (k=5,d=2,H=W=104) of y/w1, /count; 3x3 reflect smoothing; unfold*w1
// ---------------------------------------------------------------------------
__global__ void fold1_kernel(const float* __restrict__ ypre,
                             const float* __restrict__ w1, float* __restrict__ F)
{
    const int TOT = IPN * 3 * 104 * 104;
    int idx = blockIdx.x * blockDim.x + threadIdx.x;
    if (idx >= TOT) return;
    const int x = idx % 104;
    const int y = (idx / 104) % 104;
    const int c = (idx / (104 * 104)) % 3;
    const int i = idx / (3 * 104 * 104);
    const int v = y - 95, u = x - 95;
    const int ky0 = (v <= 0) ? 0 : ((v + 1) >> 1), ky1 = min(4, y >> 1);
    const int kx0 = (u <= 0) ? 0 : ((u + 1) >> 1), kx1 = min(4, x >> 1);
    float s = 0.f;
    for (int ky = ky0; ky <= ky1; ++ky)
        for (int kx = kx0; kx <= kx1; ++kx) {
            const int p = (y - 2 * ky) * 96 + (x - 2 * kx);
            s += ypre[((size_t)i * PPN + p) * PNE + c * 25 + ky * 5 + kx]
                 / w1[i * PPN + p];
        }
    F[idx] = s / (float)((ky1 - ky0 + 1) * (kx1 - kx0 + 1));
}

__global__ void smooth_kernel(const float* __restrict__ F, float* __restrict__ S)
{
    const int TOT = IPN * 3 * 104 * 104;
    int idx = blockIdx.x * blockDim.x + threadIdx.x;
    if (idx >= TOT) return;
    const int x  = idx % 104;
    const int y  = (idx / 104) % 104;
    const int cc = idx / (104 * 104);        // combined i*3 + c
    const float k[3] = {0.25f, 0.5f, 0.25f};
    float s = 0.f;
    for (int dy = -1; dy <= 1; ++dy) {
        int yy = y + dy; yy = yy < 0 ? -yy : (yy > 103 ? 206 - yy : yy);
        for (int dx = -1; dx <= 1; ++dx) {
            int xx = x + dx; xx = xx < 0 ? -xx : (xx > 103 ? 206 - xx : xx);
            s += k[dy + 1] * k[dx + 1] * F[(cc * 104 + yy) * 104 + xx];
        }
    }
    S[idx] = s;
}

__global__ void unfold1_kernel(const float* __restrict__ S,
                               const float* __restrict__ w1, float* __restrict__ yagg)
{
    const int TOT = IPN * PPN * PNE;
    int idx = blockIdx.x * blockDim.x + threadIdx.x;
    if (idx >= TOT) return;
    const int e = idx % PNE;
    const int p = (idx / PNE) % PPN;
    const int i = idx / (PNE * PPN);
    const int c = e / 25, ky = (e / 5) % 5, kx = e % 5;
    const int py = p / 96, px = p % 96;
    yagg[idx] = S[((i * 3 + c) * 104 + py + 2 * ky) * 104 + px + 2 * kx]
                * w1[i * PPN + p];
}

// ---------------------------------------------------------------------------
// Host driver
// ---------------------------------------------------------------------------
static inline dim3 blocks_for(size_t n) { return dim3((unsigned)((n + 255) / 256)); }

extern "C" void kernel_launch(void* const* d_in, const int* in_sizes, int n_in,
                              void* d_out, int out_size, void* d_ws, size_t ws_size,
                              hipStream_t stream)
{
    (void)in_sizes; (void)n_in; (void)out_size; (void)ws_size;

    // ---- inputs (top-level dict in insertion order; params pytree leaves
    //      flattened in sorted-key order, matching jax dict flattening) ------
    const float* x0 = (const float*)d_in[0];   // [2,9216,14,75]
    const float* x1 = (const float*)d_in[1];   // [2,9216,14,75]
    const float* w1 = (const float*)d_in[2];   // [2,9216,1,1]
    const float* a0po_Wh = (const float*)d_in[3];   // agg0_post {Wh[14,1],Wv[80,75],b[14,80]}
    const float* a0po_Wv = (const float*)d_in[4];
    const float* a0po_b  = (const float*)d_in[5];
    const float* a0pr_Wh = (const float*)d_in[6];   // agg0_pre {Wh[1,14],Wv[75,80],b[1,75]}
    const float* a0pr_Wv = (const float*)d_in[7];
    const float* a0pr_b  = (const float*)d_in[8];
    const float* a1po_Wh = (const float*)d_in[9];
    const float* a1po_Wv = (const float*)d_in[10];
    const float* a1po_b  = (const float*)d_in[11];
    const float* a1pr_Wh = (const float*)d_in[12];
    const float* a1pr_Wv = (const float*)d_in[13];
    const float* a1pr_b  = (const float*)d_in[14];
    // p2: sorted keys Wh2,Wh3,Who,Wv2,Wv3,Wvo,a,b2,b3,be2,be3,bo,g2,g3
    const float* p2_Wh2 = (const float*)d_in[15];
    const float* p2_Wh3 = (const float*)d_in[16];
    const float* p2_Who = (const float*)d_in[17];
    const float* p2_Wv2 = (const float*)d_in[18];
    const float* p2_Wv3 = (const float*)d_in[19];
    const float* p2_Wvo = (const float*)d_in[20];
    const float* p2_a   = (const float*)d_in[21];
    const float* p2_b2  = (const float*)d_in[22];
    const float* p2_b3  = (const float*)d_in[23];
    const float* p2_be2 = (const float*)d_in[24];
    const float* p2_be3 = (const float*)d_in[25];
    const float* p2_bo  = (const float*)d_in[26];
    const float* p2_g2  = (const float*)d_in[27];
    const float* p2_g3  = (const float*)d_in[28];
    // s0 / s1: sorted keys Wh0,Wh1,Wv0,Wv1,a,b0,b1,be1,g1
    const float* s0_Wh0 = (const float*)d_in[29];
    const float* s0_Wh1 = (const float*)d_in[30];
    const float* s0_Wv0 = (const float*)d_in[31];
    const float* s0_Wv1 = (const float*)d_in[32];
    const float* s0_a   = (const float*)d_in[33];
    const float* s0_b0  = (const float*)d_in[34];
    const float* s0_b1  = (const float*)d_in[35];
    const float* s0_be1 = (const float*)d_in[36];
    const float* s0_g1  = (const float*)d_in[37];
    const float* s1_Wh0 = (const float*)d_in[38];
    const float* s1_Wh1 = (const float*)d_in[39];
    const float* s1_Wv0 = (const float*)d_in[40];
    const float* s1_Wv1 = (const float*)d_in[41];
    const float* s1_a   = (const float*)d_in[42];
    const float* s1_b0  = (const float*)d_in[43];
    const float* s1_b1  = (const float*)d_in[44];
    const float* s1_be1 = (const float*)d_in[45];
    const float* s1_g1  = (const float*)d_in[46];

    float* out = (float*)d_out;                // [2,9216,1,75]

    // ---- workspace layout (floats) ---------------------------------------
    float* W = (float*)d_ws;
    size_t off = 0;
    float* cat   = W + off; off += (size_t)IPT * 56 * VSZ;   // concat/h/z buffer
    float* tA    = W + off; off += (size_t)IPT * 56 * VSZ;   // t scratch
    float* tB    = W + off; off += (size_t)IPT * 14 * VSZ;   // sep t scratch
    float* y0pre = W + off; off += (size_t)IPT * PNE;
    float* y1pre = W + off; off += (size_t)IPT * PNE;
    float* y0agg = W + off; off += (size_t)IPT * PNE;
    float* y1agg = W + off; off += (size_t)IPT * PNE;
    float* F0    = W + off; off += (size_t)IPN * 3 * 100 * 100;
    float* F1    = W + off; off += (size_t)IPN * 3 * 104 * 104;
    float* S1    = W + off; off += (size_t)IPN * 3 * 104 * 104;
    float* stats = W + off; off += 2 * 56;
    float* bnS   = W + off; off += 56;
    float* bnB   = W + off; off += 56;

    const float invCnt = 1.f / ((float)IPT * (float)VSZ);    // 1/(2*9216*80)
    const size_t sepN  = (size_t)IPT * 14 * VSZ;
    const size_t bigN  = (size_t)IPT * 56 * VSZ;
    const size_t preN  = (size_t)IPT * PNE;
    const int fold0N   = IPN * 3 * 100 * 100;
    const int fold1N   = IPN * 3 * 104 * 104;

    // ======================= sep_part1(x0) -> cat[:,0:14,:] ================
    verhor_t<14, PNE, VSZ, 14, 1><<<IPT, 256, 0, stream>>>(
        x0, 14, 0, s0_Wv0, s0_Wh0, s0_b0, tA, 14, 0);
    verhor_t<14, VSZ, VSZ, 14, 0><<<IPT, 256, 0, stream>>>(
        tA, 14, 0, s0_Wv1, s0_Wh1, s0_b1, tB, 14, 0);
    zero_kernel<<<1, 128, 0, stream>>>(stats, 2 * 14);
    bn_stats_kernel<<<dim3(14, 128), 256, 0, stream>>>(tB, 14, VSZ, stats);
    bn_finalize_kernel<<<1, 64, 0, stream>>>(stats, s0_g1, s0_be1, 14, invCnt, bnS, bnB);
    combine_kernel<<<blocks_for(sepN), 256, 0, stream>>>(tA, tB, bnS, bnB, s0_a,
        14, VSZ, cat, 56, 0, sepN);

    // ======================= sep_part1(x1) -> cat[:,14:28,:] ===============
    verhor_t<14, PNE, VSZ, 14, 1><<<IPT, 256, 0, stream>>>(
        x1, 14, 0, s1_Wv0, s1_Wh0, s1_b0, tA, 14, 0);
    verhor_t<14, VSZ, VSZ, 14, 0><<<IPT, 256, 0, stream>>>(
        tA, 14, 0, s1_Wv1, s1_Wh1, s1_b1, tB, 14, 0);
    zero_kernel<<<1, 128, 0, stream>>>(stats, 2 * 14);
    bn_stats_kernel<<<dim3(14, 128), 256, 0, stream>>>(tB, 14, VSZ, stats);
    bn_finalize_kernel<<<1, 64, 0, stream>>>(stats, s1_g1, s1_be1, 14, invCnt, bnS, bnB);
    combine_kernel<<<blocks_for(sepN), 256, 0, stream>>>(tA, tB, bnS, bnB, s1_a,
        14, VSZ, cat, 56, 14, sepN);

    // ======================= agg0 branch -> cat[:,28:42,:] =================
    verhor_t<14, VSZ, PNE, 1, 0><<<IPT, 256, 0, stream>>>(
        cat, 56, 0, a0pr_Wv, a0pr_Wh, a0pr_b, y0pre, 1, 0);
    fold0_kernel<<<blocks_for(fold0N), 256, 0, stream>>>(y0pre, F0);
    unfold0_kernel<<<blocks_for(preN), 256, 0, stream>>>(F0, y0agg);
    verhor_t<1, PNE, VSZ, 14, 1><<<IPT, 256, 0, stream>>>(
        y0agg, 1, 0, a0po_Wv, a0po_Wh, a0po_b, cat, 56, 28);

    // ======================= agg1 branch -> cat[:,42:56,:] =================
    verhor_t<14, VSZ, PNE, 1, 0><<<IPT, 256, 0, stream>>>(
        cat, 56, 14, a1pr_Wv, a1pr_Wh, a1pr_b, y1pre, 1, 0);
    fold1_kernel<<<blocks_for(fold1N), 256, 0, stream>>>(y1pre, w1, F1);
    smooth_kernel<<<blocks_for(fold1N), 256, 0, stream>>>(F1, S1);
    unfold1_kernel<<<blocks_for(preN), 256, 0, stream>>>(S1, w1, y1agg);
    verhor_t<1, PNE, VSZ, 14, 1><<<IPT, 256, 0, stream>>>(
        y1agg, 1, 0, a1po_Wv, a1po_Wh, a1po_b, cat, 56, 42);

    // ======================= part2 =========================================
    // t2 = verhor(cat) -> tA ; h = relu(bn(t2)) -> cat (cat fully consumed)
    verhor_t<56, VSZ, VSZ, 56, 0><<<IPT, 256, 0, stream>>>(
        cat, 56, 0, p2_Wv2, p2_Wh2, p2_b2, tA, 56, 0);
    zero_kernel<<<1, 128, 0, stream>>>(stats, 2 * 56);
    bn_stats_kernel<<<dim3(56, 128), 256, 0, stream>>>(tA, 56, VSZ, stats);
    bn_finalize_kernel<<<1, 64, 0, stream>>>(stats, p2_g2, p2_be2, 56, invCnt, bnS, bnB);
    bnrelu_kernel<<<blocks_for(bigN), 256, 0, stream>>>(tA, bnS, bnB, 56, VSZ, cat, bigN);

    // t3 = verhor(h=cat) -> tA ; z = a*h + (1-a)*relu(bn(t3)) -> cat
    verhor_t<56, VSZ, VSZ, 56, 0><<<IPT, 256, 0, stream>>>(
        cat, 56, 0, p2_Wv3, p2_Wh3, p2_b3, tA, 56, 0);
    zero_kernel<<<1, 128, 0, stream>>>(stats, 2 * 56);
    bn_stats_kernel<<<dim3(56, 128), 256, 0, stream>>>(tA, 56, VSZ, stats);
    bn_finalize_kernel<<<1, 64, 0, stream>>>(stats, p2_g3, p2_be3, 56, invCnt, bnS, bnB);
    combine_kernel<<<blocks_for(bigN), 256, 0, stream>>>(cat, tA, bnS, bnB, p2_a,
        56, VSZ, cat, 56, 0, bigN);

    // out = verhor(z=cat, Wvo[75,80], Who[1,56], bo) -> [2,9216,1,75]
    verhor_t<56, VSZ, PNE, 1, 0><<<IPT, 256, 0, stream>>>(
        cat, 56, 0, p2_Wvo, p2_Who, p2_bo, out, 1, 0);
}